// WidthAP_22393959481954
// MI455X (gfx1250) — compile-verified
//
#include <hip/hip_runtime.h>

// Problem constants (from the reference)
#define BATCH   32
#define LENGTH  2048
#define WIDTH   5
#define HDIM    768
#define PADDED  (LENGTH + WIDTH - 1)   // 2052

// Tiling
#define TILE_ROWS     64                   // output rows per block
#define HALO          (WIDTH - 1)          // 4
#define LD_ROWS       (TILE_ROWS + HALO)   // 68 rows staged in LDS
#define TILE_COLS     128                  // floats along H per block (32 lanes x float4)
#define ROWS_PER_WARP 8                    // 8 warps x 8 rows = 64 rows

typedef unsigned int       u32;
typedef unsigned long long u64;
typedef u32   u32x4 __attribute__((ext_vector_type(4)));
typedef int   i32x4 __attribute__((ext_vector_type(4)));
typedef int   i32x8 __attribute__((ext_vector_type(8)));
typedef float f32x4 __attribute__((ext_vector_type(4)));

__global__ __launch_bounds__(256) void widthap_conv_kernel(
    const float* __restrict__ x,
    const float* __restrict__ attn,
    float* __restrict__ out)
{
    __shared__ __align__(16) float tile[LD_ROWS * TILE_COLS]; // 34,816 B
    __shared__ float a_s[LD_ROWS];

    const int hc = blockIdx.x;            // 0..5   : H chunk of 128 floats
    const int it = blockIdx.y;            // 0..31  : i tile of 64 rows
    const int b  = blockIdx.z;            // 0..31  : batch
    const int i0 = it * TILE_ROWS;

    const int tid  = threadIdx.x;
    const int lane = tid & 31;
    const int warp = tid >> 5;

    // Stage the 68-entry attention slice to LDS (tiny, through normal path).
    if (tid < LD_ROWS)
        a_s[tid] = attn[b * PADDED + i0 + tid];

    // ---- Async tensor DMA: 68 x 128 fp32 tile of x -> LDS -------------------
    const u64 gaddr = (u64)(const void*)
        (x + ((u64)b * PADDED + (u64)i0) * HDIM + hc * TILE_COLS);

#if __has_builtin(__builtin_amdgcn_tensor_load_to_lds)
    if (warp == 0) {
        // D# group 0 (128b): count=1 | lds_addr | global_addr[56:0] | type=2
        u32x4 g0;
        g0[0] = 1u;                                   // count=1, user mode
        g0[1] = (u32)(uintptr_t)&tile[0];             // LDS byte address (low 32b of flat)
        g0[2] = (u32)(gaddr & 0xFFFFFFFFull);
        g0[3] = (u32)((gaddr >> 32) & 0x1FFFFFFull) | (2u << 30); // type=2 ("image")

        // D# group 1 (256b)
        i32x8 g1;
        g1[0] = (int)(2u << 16);                      // data_size=2 -> 4 bytes; mask=0
        g1[1] = (int)((u32)TILE_COLS << 16);          // tensor_dim0[15:0]  (bits 63:48)
        g1[2] = (int)((u32)LD_ROWS   << 16);          // tensor_dim0 hi=0 | tensor_dim1 lo
        g1[3] = (int)((u32)TILE_COLS << 16);          // tensor_dim1 hi=0 | tile_dim0
        g1[4] = (int)LD_ROWS;                         // tile_dim1 | tile_dim2=0
        g1[5] = (int)HDIM;                            // tensor_dim0_stride = 768 elements
        g1[6] = 0;                                    // stride0 hi | stride1 lo (unused, 2D)
        g1[7] = 0;

        i32x4 z4 = {0, 0, 0, 0};                      // groups 2/3 unused (tile_dim2=0)
        i32x8 z8 = {0, 0, 0, 0, 0, 0, 0, 0};          // extra descriptor group (unused)
        // clang-23 / therock-10.0 form: 6 args (g0, g1, g2, g3, extra, cpol)
        __builtin_amdgcn_tensor_load_to_lds(g0, g1, z4, z4, z8, /*cpol=*/0);
#if __has_builtin(__builtin_amdgcn_s_wait_tensorcnt)
        __builtin_amdgcn_s_wait_tensorcnt(0);
#endif
    }
#else
    // Fallback: cooperative b128 copy (still exactly-once global reads).
    for (int idx = tid; idx < LD_ROWS * (TILE_COLS / 4); idx += 256) {
        const int row = idx >> 5, c4 = idx & 31;
        ((f32x4*)tile)[idx] = *(const f32x4*)
            (x + ((u64)b * PADDED + (u64)(i0 + row)) * HDIM + hc * TILE_COLS + c4 * 4);
    }
#endif
    __syncthreads();

    // ---- Compute: register sliding window over LDS rows ---------------------
    const f32x4* tf4 = (const f32x4*)tile;            // row stride = 32 float4
    const int r0 = warp * ROWS_PER_WARP;

    f32x4 w0 = tf4[(r0 + 0) * 32 + lane];
    f32x4 w1 = tf4[(r0 + 1) * 32 + lane];
    f32x4 w2 = tf4[(r0 + 2) * 32 + lane];
    f32x4 w3 = tf4[(r0 + 3) * 32 + lane];
    float a0 = a_s[r0 + 0], a1 = a_s[r0 + 1], a2 = a_s[r0 + 2], a3 = a_s[r0 + 3];

    float* op = out + ((u64)b * LENGTH + (u64)(i0 + r0)) * HDIM
                    + hc * TILE_COLS + lane * 4;

#pragma unroll
    for (int k = 0; k < ROWS_PER_WARP; ++k) {
        f32x4 w4 = tf4[(r0 + k + 4) * 32 + lane];
        float a4 = a_s[r0 + k + 4];

        f32x4 acc = w0 * a0 + w1 * a1 + w2 * a2 + w3 * a3 + w4 * a4; // contracts to FMA

        __builtin_nontemporal_store(acc, (f32x4*)op);  // write-once stream, NT hint
        op += HDIM;

        w0 = w1; w1 = w2; w2 = w3; w3 = w4;
        a0 = a1; a1 = a2; a2 = a3; a3 = a4;
    }
}

extern "C" void kernel_launch(void* const* d_in, const int* in_sizes, int n_in,
                              void* d_out, int out_size, void* d_ws, size_t ws_size,
                              hipStream_t stream) {
    const float* x    = (const float*)d_in[0];   // (32, 1, 2052, 768) fp32
    const float* attn = (const float*)d_in[1];   // (32, 2052) fp32
    float* out        = (float*)d_out;           // (32, 1, 2048, 768) fp32

    dim3 grid(HDIM / TILE_COLS, LENGTH / TILE_ROWS, BATCH);  // (6, 32, 32)
    widthap_conv_kernel<<<grid, 256, 0, stream>>>(x, attn, out);
}